// PROG_RNN_DET_26362509263310
// MI455X (gfx1250) — compile-verified
//
#include <hip/hip_runtime.h>

// ---------------------------------------------------------------------------
// PROG_RNN_DET (macro): persistent-kernel formulation.
// The GRU recurrence couples rows only to themselves, so each block owns 16
// batch rows and runs all 255 timesteps with h0/h1/d1/d2 resident in LDS.
// fp32 tensor path: V_WMMA_F32_16X16X4_F32. 2 launches total.
// ---------------------------------------------------------------------------

#define TT   256
#define BB   2048
#define YY   20
#define HH   256
#define LDP  260          // LDS row stride (floats): bank-conflict-free frags

typedef __attribute__((ext_vector_type(2))) float v2f;
typedef __attribute__((ext_vector_type(8))) float v8f;

__device__ __forceinline__ v8f wmma4(v2f a, v2f b, v8f c) {
  return __builtin_amdgcn_wmma_f32_16x16x4_f32(false, a, false, b, (short)0, c,
                                               false, false);
}

// Global fragment load (A: activations [rows, ld]; B: weights [out, K]).
__device__ __forceinline__ v2f frag_ld(const float* __restrict__ base,
                                       int row0, int ld, int k, int lane) {
  int r  = row0 + (lane & 15);
  int kk = k + ((lane >> 4) << 1);
  const float* p = base + (size_t)r * ld + kk;
  v2f out = {p[0], p[1]};
  return out;
}

// LDS A-fragment: row = lane&15, k-pair by half-wave. Stride 260 -> bank 4m.
__device__ __forceinline__ v2f lds_frag(const float* s, int k, int lane) {
  int kk = k + ((lane >> 4) << 1);
  const float* p = s + (lane & 15) * LDP + kk;
  v2f out = {p[0], p[1]};
  return out;
}

__device__ __forceinline__ float fsigmoid(float x) {
  return 1.0f / (1.0f + __expf(-x));
}
__device__ __forceinline__ float ftanh(float x) {
  float sg = (x >= 0.0f) ? 1.0f : -1.0f;
  float e  = __expf(-2.0f * fabsf(x));
  return sg * (1.0f - e) / (1.0f + e);
}

// ---------------------------------------------------------------------------
// Persistent kernel: 128 blocks x 256 threads (8 waves). Block owns rows
// [16*bid, 16*bid+16). Wave w owns column blocks {w, w+8} of each 256-wide
// activation. Loss accumulated in registers across all steps.
// ---------------------------------------------------------------------------
__global__ void __launch_bounds__(256)
prog_rnn_persistent(const float* __restrict__ data,
                    const float* __restrict__ Wih0, const float* __restrict__ Whh0,
                    const float* __restrict__ bih0, const float* __restrict__ bhh0,
                    const float* __restrict__ Wih1, const float* __restrict__ Whh1,
                    const float* __restrict__ bih1, const float* __restrict__ bhh1,
                    const float* __restrict__ W1, const float* __restrict__ b1,
                    const float* __restrict__ W2, const float* __restrict__ b2,
                    const float* __restrict__ W3, const float* __restrict__ b3,
                    float* __restrict__ lossPartial) {
  __shared__ float h0s[16 * LDP];
  __shared__ float h1s[16 * LDP];
  __shared__ float d1s[16 * LDP];
  __shared__ float d2s[16 * LDP];
  __shared__ float lred[8];

  const int tid  = threadIdx.x;
  const int lane = tid & 31;
  const int w    = tid >> 5;        // wave 0..7
  const int r0   = blockIdx.x << 4; // global row base (16 rows per block)
  const int nl   = lane & 15;
  const int rhi  = (lane >> 4) << 3;
  const int n0[2] = {w << 4, (w + 8) << 4};
  const int nn_[2] = {n0[0] + nl, n0[1] + nl};

  // h(t=0) = 0
  for (int i = tid; i < 16 * LDP; i += 256) { h0s[i] = 0.0f; h1s[i] = 0.0f; }
  __syncthreads();

  // Hoist per-column biases (constant over time).
  float b0r[2], b0z[2], b0n[2], c0r[2], c0z[2], c0n[2];
  float b1r[2], b1z[2], b1n[2], c1r[2], c1z[2], c1n[2];
  float bd1[2], bd2[2];
#pragma unroll
  for (int s = 0; s < 2; ++s) {
    int n = nn_[s];
    b0r[s] = bih0[n]; b0z[s] = bih0[256 + n]; b0n[s] = bih0[512 + n];
    c0r[s] = bhh0[n]; c0z[s] = bhh0[256 + n]; c0n[s] = bhh0[512 + n];
    b1r[s] = bih1[n]; b1z[s] = bih1[256 + n]; b1n[s] = bih1[512 + n];
    c1r[s] = bhh1[n]; c1z[s] = bhh1[256 + n]; c1n[s] = bhh1[512 + n];
    bd1[s] = b1[n];   bd2[s] = b2[n];
  }
  const int n3 = (w << 4) + nl;       // stage-3 column (waves 0,1 only)
  float b3v = (w < 2 && n3 < YY) ? b3[n3] : 0.0f;

  float lsum = 0.0f;

  for (int t = 0; t < TT - 1; ++t) {
    const float* xt = data + (size_t)t * (BB * YY);

    // ---------------- GRU layer 0 ----------------
    {
      v8f acc[2][6];
      v8f z8 = {};
#pragma unroll
      for (int s = 0; s < 2; ++s)
#pragma unroll
        for (int g = 0; g < 6; ++g) acc[s][g] = z8;

      for (int k4 = 0; k4 < YY / 4; ++k4) {        // gx: x_t @ Wih0.T, K=20
        int k = k4 << 2;
        v2f a = frag_ld(xt, r0, YY, k, lane);
#pragma unroll
        for (int g = 0; g < 3; ++g) {
          v2f bA = frag_ld(Wih0 + (size_t)g * 256 * YY, n0[0], YY, k, lane);
          v2f bB = frag_ld(Wih0 + (size_t)g * 256 * YY, n0[1], YY, k, lane);
          acc[0][g] = wmma4(a, bA, acc[0][g]);
          acc[1][g] = wmma4(a, bB, acc[1][g]);
        }
      }
      for (int k4 = 0; k4 < 64; ++k4) {            // gh: h0 @ Whh0.T, K=256
        int k = k4 << 2;
        v2f a = lds_frag(h0s, k, lane);
#pragma unroll
        for (int g = 0; g < 3; ++g) {
          v2f bA = frag_ld(Whh0 + (size_t)g * 256 * HH, n0[0], HH, k, lane);
          v2f bB = frag_ld(Whh0 + (size_t)g * 256 * HH, n0[1], HH, k, lane);
          acc[0][3 + g] = wmma4(a, bA, acc[0][3 + g]);
          acc[1][3 + g] = wmma4(a, bB, acc[1][3 + g]);
        }
      }
      float hnew[2][8];
#pragma unroll
      for (int s = 0; s < 2; ++s) {
        int n = nn_[s];
#pragma unroll
        for (int i = 0; i < 8; ++i) {
          float r  = fsigmoid(acc[s][0][i] + b0r[s] + acc[s][3][i] + c0r[s]);
          float zz = fsigmoid(acc[s][1][i] + b0z[s] + acc[s][4][i] + c0z[s]);
          float nv = ftanh(acc[s][2][i] + b0n[s] + r * (acc[s][5][i] + c0n[s]));
          float hp = h0s[(i + rhi) * LDP + n];
          hnew[s][i] = (1.0f - zz) * nv + zz * hp;
        }
      }
      __syncthreads();
#pragma unroll
      for (int s = 0; s < 2; ++s)
#pragma unroll
        for (int i = 0; i < 8; ++i)
          h0s[(i + rhi) * LDP + nn_[s]] = hnew[s][i];
      __syncthreads();
    }

    // ---------------- GRU layer 1 ----------------
    {
      v8f acc[2][6];
      v8f z8 = {};
#pragma unroll
      for (int s = 0; s < 2; ++s)
#pragma unroll
        for (int g = 0; g < 6; ++g) acc[s][g] = z8;

      for (int k4 = 0; k4 < 64; ++k4) {            // gx: h0_new @ Wih1.T
        int k = k4 << 2;
        v2f a = lds_frag(h0s, k, lane);
#pragma unroll
        for (int g = 0; g < 3; ++g) {
          v2f bA = frag_ld(Wih1 + (size_t)g * 256 * HH, n0[0], HH, k, lane);
          v2f bB = frag_ld(Wih1 + (size_t)g * 256 * HH, n0[1], HH, k, lane);
          acc[0][g] = wmma4(a, bA, acc[0][g]);
          acc[1][g] = wmma4(a, bB, acc[1][g]);
        }
      }
      for (int k4 = 0; k4 < 64; ++k4) {            // gh: h1 @ Whh1.T
        int k = k4 << 2;
        v2f a = lds_frag(h1s, k, lane);
#pragma unroll
        for (int g = 0; g < 3; ++g) {
          v2f bA = frag_ld(Whh1 + (size_t)g * 256 * HH, n0[0], HH, k, lane);
          v2f bB = frag_ld(Whh1 + (size_t)g * 256 * HH, n0[1], HH, k, lane);
          acc[0][3 + g] = wmma4(a, bA, acc[0][3 + g]);
          acc[1][3 + g] = wmma4(a, bB, acc[1][3 + g]);
        }
      }
      float hnew[2][8];
#pragma unroll
      for (int s = 0; s < 2; ++s) {
        int n = nn_[s];
#pragma unroll
        for (int i = 0; i < 8; ++i) {
          float r  = fsigmoid(acc[s][0][i] + b1r[s] + acc[s][3][i] + c1r[s]);
          float zz = fsigmoid(acc[s][1][i] + b1z[s] + acc[s][4][i] + c1z[s]);
          float nv = ftanh(acc[s][2][i] + b1n[s] + r * (acc[s][5][i] + c1n[s]));
          float hp = h1s[(i + rhi) * LDP + n];
          hnew[s][i] = (1.0f - zz) * nv + zz * hp;
        }
      }
      __syncthreads();
#pragma unroll
      for (int s = 0; s < 2; ++s)
#pragma unroll
        for (int i = 0; i < 8; ++i)
          h1s[(i + rhi) * LDP + nn_[s]] = hnew[s][i];
      __syncthreads();
    }

    // ---------------- Decoder stage 1: d1 = relu(h1 @ W1.T + b1) ----------
    {
      v8f acc[2];
      v8f z8 = {};
      acc[0] = z8; acc[1] = z8;
      for (int k4 = 0; k4 < 64; ++k4) {
        int k = k4 << 2;
        v2f a = lds_frag(h1s, k, lane);
        acc[0] = wmma4(a, frag_ld(W1, n0[0], HH, k, lane), acc[0]);
        acc[1] = wmma4(a, frag_ld(W1, n0[1], HH, k, lane), acc[1]);
      }
#pragma unroll
      for (int s = 0; s < 2; ++s)
#pragma unroll
        for (int i = 0; i < 8; ++i) {
          float v = acc[s][i] + bd1[s];
          d1s[(i + rhi) * LDP + nn_[s]] = v > 0.0f ? v : 0.0f;
        }
      __syncthreads();
    }
    // ---------------- Decoder stage 2: d2 = relu(d1 @ W2.T + b2) ----------
    {
      v8f acc[2];
      v8f z8 = {};
      acc[0] = z8; acc[1] = z8;
      for (int k4 = 0; k4 < 64; ++k4) {
        int k = k4 << 2;
        v2f a = lds_frag(d1s, k, lane);
        acc[0] = wmma4(a, frag_ld(W2, n0[0], HH, k, lane), acc[0]);
        acc[1] = wmma4(a, frag_ld(W2, n0[1], HH, k, lane), acc[1]);
      }
#pragma unroll
      for (int s = 0; s < 2; ++s)
#pragma unroll
        for (int i = 0; i < 8; ++i) {
          float v = acc[s][i] + bd2[s];
          d2s[(i + rhi) * LDP + nn_[s]] = v > 0.0f ? v : 0.0f;
        }
      __syncthreads();
    }
    // ------- Decoder stage 3 (waves 0,1): d3 = d2 @ W3.T + b3, SSE --------
    if (w < 2) {                       // wave-uniform branch: EXEC stays full
      v8f acc3 = {};
      for (int k4 = 0; k4 < 64; ++k4) {
        int kk = (k4 << 2) + ((lane >> 4) << 1);
        v2f a = lds_frag(d2s, k4 << 2, lane);
        v2f b;
        if (n3 < YY) {
          const float* q = W3 + (size_t)n3 * HH + kk;
          b.x = q[0]; b.y = q[1];
        } else {
          b.x = 0.0f; b.y = 0.0f;
        }
        acc3 = wmma4(a, b, acc3);
      }
      if (n3 < YY) {
        const float* nxt = data + (size_t)(t + 1) * (BB * YY);
#pragma unroll
        for (int i = 0; i < 8; ++i) {
          float d = acc3[i] + b3v - nxt[(size_t)(r0 + i + rhi) * YY + n3];
          lsum += d * d;
        }
      }
    }
    __syncthreads();
  }

  // Deterministic reduction: wave shuffle tree, then wave0 sums 8 partials.
#pragma unroll
  for (int off = 16; off > 0; off >>= 1)
    lsum += __shfl_xor(lsum, off, 32);
  if (lane == 0) lred[w] = lsum;
  __syncthreads();
  if (tid == 0) {
    float s = 0.0f;
#pragma unroll
    for (int i = 0; i < 8; ++i) s += lred[i];
    lossPartial[blockIdx.x] = s;
  }
}

// Fixed-order tree reduction of 128 block partials -> d_out[0]
__global__ void __launch_bounds__(128)
loss_reduce(const float* __restrict__ part, float* __restrict__ out) {
  __shared__ float s[128];
  s[threadIdx.x] = part[threadIdx.x];
  __syncthreads();
  for (int off = 64; off > 0; off >>= 1) {
    if ((int)threadIdx.x < off) s[threadIdx.x] += s[threadIdx.x + off];
    __syncthreads();
  }
  if (threadIdx.x == 0) out[0] = s[0];
}

extern "C" void kernel_launch(void* const* d_in, const int* in_sizes, int n_in,
                              void* d_out, int out_size, void* d_ws, size_t ws_size,
                              hipStream_t stream) {
  const float* data = (const float*)d_in[0];
  const float* Wih0 = (const float*)d_in[1];
  const float* Whh0 = (const float*)d_in[2];
  const float* bih0 = (const float*)d_in[3];
  const float* bhh0 = (const float*)d_in[4];
  const float* Wih1 = (const float*)d_in[5];
  const float* Whh1 = (const float*)d_in[6];
  const float* bih1 = (const float*)d_in[7];
  const float* bhh1 = (const float*)d_in[8];
  const float* W1   = (const float*)d_in[9];
  const float* b1   = (const float*)d_in[10];
  const float* W2   = (const float*)d_in[11];
  const float* b2   = (const float*)d_in[12];
  const float* W3   = (const float*)d_in[13];
  const float* b3   = (const float*)d_in[14];

  float* lossP = (float*)d_ws;   // 128 floats

  prog_rnn_persistent<<<BB / 16, 256, 0, stream>>>(
      data, Wih0, Whh0, bih0, bhh0, Wih1, Whh1, bih1, bhh1,
      W1, b1, W2, b2, W3, b3, lossP);
  loss_reduce<<<1, 128, 0, stream>>>(lossP, (float*)d_out);
}